// Pooling_36799279792878
// MI455X (gfx1250) — compile-verified
//
#include <hip/hip_runtime.h>
#include <math.h>

// x: [N=512, 3, 3, NUM=128, FRAME=100] f32  (frame innermost / contiguous)
// y: [N, 3, 3, NUM/2=64, FRAME=100]    f32
//
// Pure streaming select/gather: memory-bound (~275 MB moved -> ~12us at
// 23.3 TB/s). No matrix product exists in this op, so WMMA is not used;
// the CDNA5-specific optimization is NT temporal hints on the one-pass
// 354 MB stream (working set >> 192 MB L2) plus wave32-coalesced rows.

#define NN      512
#define NUMM    128
#define PAIRS   64          // NUM/2
#define FRAMES  100
#define IJ_STRIDE_X  (NUMM * FRAMES)    // 12800
#define IJ_STRIDE_Y  (PAIRS * FRAMES)   // 6400

__device__ __forceinline__ float rot_angle(float tr) {
    // Faithful to reference masking:
    //   maskpi   = (|tr+1| > eps) & (|tr-3| <= eps)  -> tr == 3 (f32 exact)
    //   maskacos = (|tr+1| <= eps)                   -> tr == -1 (f32 exact)
    //   r = tr*pi*maskpi + acos(tr)*maskacos
    // For in-domain tr (guaranteed by harness) the gated-branch form below is
    // bit-identical to the multiply-by-mask form.
    const float eps = 1e-12f;
    const float apl = fabsf(tr + 1.0f);
    float r = 0.0f;
    if ((apl > eps) && (fabsf(tr - 3.0f) <= eps))
        r += tr * 3.14159265358979323846f;
    if (apl <= eps)
        r += acosf(tr);
    return r;
}

__global__ void pool_space_kernel(const float* __restrict__ x,
                                  float* __restrict__ y) {
    const int f   = threadIdx.x;          // frame index, 0..99 active of 128
    const int blk = blockIdx.x;           // n * PAIRS + p
    const int p   = blk & (PAIRS - 1);
    const int n   = blk >> 6;
    if (f >= FRAMES) return;

    const int    m0 = 2 * p;
    const size_t xn = (size_t)n * 9 * IJ_STRIDE_X;
    const size_t yn = (size_t)n * 9 * IJ_STRIDE_Y;

    // Diagonal elements of both pair members (6 NT loads) -> traces.
    float d0[3], d1[3];
#pragma unroll
    for (int i = 0; i < 3; ++i) {
        const size_t base = xn + (size_t)(i * 3 + i) * IJ_STRIDE_X
                               + (size_t)m0 * FRAMES + (size_t)f;
        d0[i] = __builtin_nontemporal_load(&x[base]);
        d1[i] = __builtin_nontemporal_load(&x[base + FRAMES]);
    }
    const float r0 = rot_angle(d0[0] + d0[1] + d0[2]);
    const float r1 = rot_angle(d1[0] + d1[1] + d1[2]);
    // jnp.argmax over [r0, r1]: first max on ties -> 1 only if strictly greater
    const int k = (r1 > r0) ? 1 : 0;

    const size_t msel = (size_t)(m0 + k) * FRAMES + (size_t)f;
    const size_t yoff = (size_t)p * FRAMES + (size_t)f;

    // Copy chosen 3x3: diagonal from registers, off-diagonal gathered (6 NT
    // loads), 9 NT stores. All accesses coalesced along frame.
#pragma unroll
    for (int i = 0; i < 3; ++i) {
#pragma unroll
        for (int j = 0; j < 3; ++j) {
            const int ij = i * 3 + j;
            float v;
            if (i == j) {
                v = k ? d1[i] : d0[i];
            } else {
                v = __builtin_nontemporal_load(
                        &x[xn + (size_t)ij * IJ_STRIDE_X + msel]);
            }
            __builtin_nontemporal_store(
                v, &y[yn + (size_t)ij * IJ_STRIDE_Y + yoff]);
        }
    }
}

extern "C" void kernel_launch(void* const* d_in, const int* in_sizes, int n_in,
                              void* d_out, int out_size, void* d_ws, size_t ws_size,
                              hipStream_t stream) {
    (void)in_sizes; (void)n_in; (void)out_size; (void)d_ws; (void)ws_size;
    const float* x = (const float*)d_in[0];
    float*       y = (float*)d_out;

    const int blocks = NN * PAIRS;   // 32768 blocks
    pool_space_kernel<<<blocks, 128, 0, stream>>>(x, y);
}